// GHMC_loss_63995012710873
// MI455X (gfx1250) — compile-verified
//
#include <hip/hip_runtime.h>

#define TPB   256
#define VEC   8
#define TILE  (TPB * VEC)   // 2048 elements per tile per workgroup
#define NBINS 10
#define HSTR  11            // padded stride for per-thread private histograms

// ---------------- async-to-LDS helpers (gfx1250) ----------------

// GVS mode: mem = SADDR(64) + VADDR(32) + IOFFSET ; VDST = wave-relative LDS byte addr.
// IOFFSET is added to BOTH the global and LDS addresses (CDNA5 ISA),
// so one set of address registers serves both 16B beats of each lane's 32B slot.
template <int OFF>
__device__ __forceinline__ void async_load_b128(unsigned ldsOff, unsigned vOff,
                                                const void* base) {
  asm volatile("global_load_async_to_lds_b128 %0, %1, %2 offset:%3"
               :: "v"(ldsOff), "v"(vOff), "s"(base), "n"(OFF)
               : "memory");
}

__device__ __forceinline__ void wait_async_le4() {
  asm volatile("s_wait_asynccnt 4" ::: "memory");
}
__device__ __forceinline__ void wait_async_0() {
  asm volatile("s_wait_asynccnt 0" ::: "memory");
}
// allow the 16 in-flight histogram ds_adds; guarantees all OLDER ds ops (the
// previous iteration's staging-buffer ds_loads) have completed -> WAR-safe reuse
__device__ __forceinline__ void wait_ds_le16() {
  asm volatile("s_wait_dscnt 16" ::: "memory");
}

// low 32 bits of a generic pointer to __shared__ == wave-relative LDS offset
__device__ __forceinline__ unsigned lds_off(const void* p) {
  return (unsigned)(unsigned long long)p;
}

// ---------------- kernels ----------------

__global__ void ghmc_init(double* gSum, unsigned* gCnt) {
  int i = threadIdx.x;
  if (i < NBINS) { gSum[i] = 0.0; gCnt[i] = 0u; }
}

__global__ __launch_bounds__(TPB) void ghmc_main(const float* __restrict__ pred,
                                                 const float* __restrict__ tgt,
                                                 double* __restrict__ gSum,
                                                 unsigned* __restrict__ gCnt,
                                                 int n) {
  __shared__ __align__(16) float    sP[2][TILE];
  __shared__ __align__(16) float    sT[2][TILE];
  __shared__ float    sSum[TPB * HSTR];
  __shared__ unsigned sCnt[TPB * HSTR];

  const int tid = threadIdx.x;
  const unsigned mb = (unsigned)tid * HSTR;

  // zero private histogram slots (this thread's only)
  #pragma unroll
  for (int b = 0; b < HSTR; ++b) { sSum[mb + b] = 0.0f; sCnt[mb + b] = 0u; }

  const unsigned vOff = (unsigned)tid * (VEC * 4u);  // 32B per lane per array
  const unsigned ldsP0 = lds_off(&sP[0][tid * VEC]);
  const unsigned ldsP1 = lds_off(&sP[1][tid * VEC]);
  const unsigned ldsT0 = lds_off(&sT[0][tid * VEC]);
  const unsigned ldsT1 = lds_off(&sT[1][tid * VEC]);

  auto elem = [&](float x, float t) {
    const float ax  = fabsf(x);
    const float em  = __expf(-ax);                      // exp(-|x|) -> v_exp_f32 path
    const float sm  = 1.0f + em;
    const float inv = __builtin_amdgcn_rcpf(sm);        // 1/(1+e^-|x|)
    const float y   = (t != 0.0f) ? -x : x;             // g = sigmoid(y)
    const float g   = (y >= 0.0f) ? inv : em * inv;
    const float l   = __logf(sm);                       // log1p(exp(-|x|))
    const float bce = fmaxf(x, 0.0f) - x * t + l;
    int b = (int)(g * 10.0f);
    if (b > 9) b = 9;
    atomicAdd(&sSum[mb + (unsigned)b], bce);            // ds_add_f32, private slot
    atomicAdd(&sCnt[mb + (unsigned)b], 1u);             // ds_add_u32, private slot
  };

  auto issue_tile = [&](int tileIdx, int buf) {
    const float* bp = pred + (size_t)tileIdx * TILE;
    const float* bt = tgt  + (size_t)tileIdx * TILE;
    const unsigned lp = buf ? ldsP1 : ldsP0;
    const unsigned lt = buf ? ldsT1 : ldsT0;
    async_load_b128<0>(lp, vOff, bp);
    async_load_b128<16>(lp, vOff, bp);
    async_load_b128<0>(lt, vOff, bt);
    async_load_b128<16>(lt, vOff, bt);
  };

  const int ntiles = n / TILE;
  const int gs = (int)gridDim.x;

  int tile = (int)blockIdx.x;
  int p = 0;
  bool have = tile < ntiles;
  if (have) issue_tile(tile, 0);

  while (have) {
    const int next = tile + gs;
    const bool hn = next < ntiles;
    if (hn) {
      wait_ds_le16();          // WAR guard for buffer p^1 (see helper comment)
      issue_tile(next, p ^ 1);
      wait_async_le4();        // current tile's 4 transfers (older, in-order) done
    } else {
      wait_async_0();
    }
    const float4 P0 = *(const float4*)&sP[p][tid * VEC];
    const float4 P1 = *(const float4*)&sP[p][tid * VEC + 4];
    const float4 T0 = *(const float4*)&sT[p][tid * VEC];
    const float4 T1 = *(const float4*)&sT[p][tid * VEC + 4];
    elem(P0.x, T0.x); elem(P0.y, T0.y); elem(P0.z, T0.z); elem(P0.w, T0.w);
    elem(P1.x, T1.x); elem(P1.y, T1.y); elem(P1.z, T1.z); elem(P1.w, T1.w);
    tile = next; p ^= 1; have = hn;
  }

  // tail (n not multiple of TILE) — plain guarded loads
  for (int i = ntiles * TILE + (int)blockIdx.x * TPB + tid; i < n; i += gs * TPB)
    elem(pred[i], tgt[i]);

  __syncthreads();

  // per-block reduction of private histograms, then global accumulation
  for (int b = tid; b < NBINS; b += TPB) {
    double ds = 0.0;
    unsigned c = 0u;
    for (int t = 0; t < TPB; ++t) {
      ds += (double)sSum[t * HSTR + b];
      c  += sCnt[t * HSTR + b];
    }
    atomicAdd(&gSum[b], ds);
    atomicAdd(&gCnt[b], c);
  }
}

__global__ void ghmc_final(const double* gSum, const unsigned* gCnt,
                           float* out, int n) {
  if (threadIdx.x == 0 && blockIdx.x == 0) {
    const double total = (double)n;
    double acc = 0.0;
    #pragma unroll
    for (int b = 0; b < NBINS; ++b) {
      unsigned c = gCnt[b];
      if (c == 0u) c = 1u;
      const double w = 0.75 * total / (double)c * 0.1;  // GHM * total / cnt / 10
      acc += w * gSum[b];
    }
    out[0] = (float)(acc / total);                      // mean
  }
}

// ---------------- host entry ----------------

extern "C" void kernel_launch(void* const* d_in, const int* in_sizes, int n_in,
                              void* d_out, int out_size, void* d_ws, size_t ws_size,
                              hipStream_t stream) {
  const float* pred = (const float*)d_in[0];
  const float* tgt  = (const float*)d_in[1];
  const int n = in_sizes[0];

  double*   gSum = (double*)d_ws;
  unsigned* gCnt = (unsigned*)((char*)d_ws + NBINS * sizeof(double));

  ghmc_init<<<1, 32, 0, stream>>>(gSum, gCnt);

  int ntiles = n / TILE;
  int grid = ntiles < 2048 ? ntiles : 2048;
  if (grid < 1) grid = 1;
  ghmc_main<<<grid, TPB, 0, stream>>>(pred, tgt, gSum, gCnt, n);

  ghmc_final<<<1, 32, 0, stream>>>(gSum, gCnt, (float*)d_out, n);
}